// BicliqueGCN_53437983097036
// MI455X (gfx1250) — compile-verified
//
#include <hip/hip_runtime.h>
#include <hip/hip_bf16.h>

// ---------------------------------------------------------------------------
// BicliqueGCN for MI455X (gfx1250, wave32).
//  - SpMM (segment_sum) via 16-lane/edge float4 gathers + global_atomic_add_f32.
//    Both layer buffers (38.4MB each) fit in the 192MB L2 simultaneously, so
//    the random gathers/scatters run at L2 bandwidth; edge streams are loaded
//    non-temporal so they don't evict the embedding working set.
//  - acc += e on the matrix pipe: D = I*E + C with 4x v_wmma_f32_16x16x4_f32
//    per 16x16 tile (exact f32 arithmetic); A/B operands staged before the
//    WMMA chain so the 4 WMMAs issue back-to-back.
// ---------------------------------------------------------------------------

typedef __attribute__((ext_vector_type(2))) float v2f;
typedef __attribute__((ext_vector_type(8))) float v8f;

#define NUM_USERS 100000
#define NUM_ITEMS 50000
#define NUM_NODES 150000
#define NUM_BQ    20000
#define EMB       64

// ---- init: e0 = out = concat(user_emb, item_emb), float4 vectorized --------
__global__ void __launch_bounds__(256) copy_init_kernel(
    const float* __restrict__ ue, const float* __restrict__ ie,
    float* __restrict__ e0, float* __restrict__ out,
    int totalVec, int userVec) {
  int g = blockIdx.x * blockDim.x + threadIdx.x;
  if (g >= totalVec) return;
  float4 v = (g < userVec) ? ((const float4*)ue)[g]
                           : ((const float4*)ie)[g - userVec];
  ((float4*)e0)[g]  = v;
  ((float4*)out)[g] = v;
}

// ---- COO SpMM: 16 lanes per edge, full 256B row per group ------------------
// Optionally fuses the weighted row-degree histogram (lane j==0).
// Edge streams (row/col/val) are touched exactly once -> non-temporal loads
// keep L2 free for the embedding buffers.
__global__ void __launch_bounds__(256) spmm16_kernel(
    const int* __restrict__ row, const int* __restrict__ col,
    const float* __restrict__ val, const float* __restrict__ x,
    float* __restrict__ y, float* __restrict__ deg, int nnz) {
  int g = blockIdx.x * blockDim.x + threadIdx.x;
  int e = g >> 4;
  int j = g & 15;
  if (e >= nnz) return;
  int   c = __builtin_nontemporal_load(col + e);
  int   r = __builtin_nontemporal_load(row + e);
  float v = __builtin_nontemporal_load(val + e);
  float4 xv = ((const float4*)(x + (size_t)c * EMB))[j];   // global_load_b128
  float* yp = y + (size_t)r * EMB + j * 4;
  unsafeAtomicAdd(yp + 0, v * xv.x);                       // global_atomic_add_f32
  unsafeAtomicAdd(yp + 1, v * xv.y);
  unsafeAtomicAdd(yp + 2, v * xv.z);
  unsafeAtomicAdd(yp + 3, v * xv.w);
  if (deg != nullptr && j == 0)
    unsafeAtomicAdd(&deg[r], v);
}

// ---- acc += e on the matrix pipe: one wave per 16x16 tile ------------------
// A(16x4 f32) layout: lanes0-15 M=lane, v0:K=0 v1:K=1; lanes16-31 v0:K=2 v1:K=3
// B(4x16) / C,D(16x16): rows striped across lanes (N = lane%16),
//   B: v0 holds rows {0,2}, v1 rows {1,3}; C/D: vr holds rows {r, r+8}.
__global__ void __launch_bounds__(128) acc_wmma_kernel(
    const float* __restrict__ e, float* __restrict__ acc) {
  int wave = threadIdx.x >> 5;          // 4 waves: one per 16-col tile
  int lane = threadIdx.x & 31;
  int n    = lane & 15;
  int hi   = lane >> 4;                 // 0: lanes 0-15, 1: lanes 16-31
  int row0 = blockIdx.x * 16;
  int col0 = wave * 16;

  const size_t base = (size_t)row0 * EMB + col0 + n;
  const float* ap = acc + base;
  const float* ep = e + base;

  // Stage ALL operands first so the 4 WMMAs can issue back-to-back.
  v8f c;
#pragma unroll
  for (int r = 0; r < 8; ++r)
    c[r] = ap[(size_t)(r + 8 * hi) * EMB];

  v2f a[4], b[4];
#pragma unroll
  for (int k = 0; k < 4; ++k) {
    a[k].x = (n == (4 * k + 2 * hi))     ? 1.0f : 0.0f;   // I[:,4k..4k+3]
    a[k].y = (n == (4 * k + 2 * hi + 1)) ? 1.0f : 0.0f;
    b[k].x = ep[(size_t)(4 * k + 2 * hi) * EMB];          // E rows 4k..4k+3
    b[k].y = ep[(size_t)(4 * k + 2 * hi + 1) * EMB];
  }

#pragma unroll
  for (int k = 0; k < 4; ++k)
    c = __builtin_amdgcn_wmma_f32_16x16x4_f32(
        false, a[k], false, b[k], (short)0, c, false, false);

  float* op = acc + base;
#pragma unroll
  for (int r = 0; r < 8; ++r)
    op[(size_t)(r + 8 * hi) * EMB] = c[r];
}

// ---- in-place row normalization: x[r,:] /= max(deg[r], !=0) ----------------
__global__ void __launch_bounds__(256) rownorm_kernel(
    float* __restrict__ x, const float* __restrict__ deg, int rowsVec) {
  int g = blockIdx.x * blockDim.x + threadIdx.x;
  if (g >= rowsVec) return;
  float d = deg[g >> 4];
  d = (d == 0.0f) ? 1.0f : d;
  float inv = 1.0f / d;
  float4 v = ((float4*)x)[g];
  v.x *= inv; v.y *= inv; v.z *= inv; v.w *= inv;
  ((float4*)x)[g] = v;
}

// ---- finalize: out = acc/4 (+ u_local/deg_hu for user rows) ----------------
__global__ void __launch_bounds__(256) finalize_kernel(
    float* __restrict__ out, const float* __restrict__ ulocal,
    const float* __restrict__ deg, int totalVec, int userVec) {
  int g = blockIdx.x * blockDim.x + threadIdx.x;
  if (g >= totalVec) return;
  float4 v = ((float4*)out)[g];
  v.x *= 0.25f; v.y *= 0.25f; v.z *= 0.25f; v.w *= 0.25f;
  if (g < userVec) {
    float d = deg[g >> 4];
    d = (d == 0.0f) ? 1.0f : d;
    float inv = 1.0f / d;
    float4 w = ((const float4*)ulocal)[g];
    v.x += w.x * inv; v.y += w.y * inv; v.z += w.z * inv; v.w += w.w * inv;
  }
  ((float4*)out)[g] = v;
}

static inline int cdiv(long long a, int b) { return (int)((a + b - 1) / b); }

extern "C" void kernel_launch(void* const* d_in, const int* in_sizes, int n_in,
                              void* d_out, int out_size, void* d_ws, size_t ws_size,
                              hipStream_t stream) {
  const float* user_emb = (const float*)d_in[0];
  const float* item_emb = (const float*)d_in[1];
  const float* adj_val  = (const float*)d_in[2];
  const float* hv_val   = (const float*)d_in[3];
  const float* hu_val   = (const float*)d_in[4];
  const int*   adj_row  = (const int*)d_in[5];
  const int*   adj_col  = (const int*)d_in[6];
  const int*   hv_row   = (const int*)d_in[7];
  const int*   hv_col   = (const int*)d_in[8];
  const int*   hu_row   = (const int*)d_in[9];
  const int*   hu_col   = (const int*)d_in[10];

  const int E_adj = in_sizes[2];
  const int E_hv  = in_sizes[3];
  const int E_hu  = in_sizes[4];

  float* out = (float*)d_out;

  // workspace layout (bytes); peak ~82.4 MB
  const size_t embBytesN = (size_t)NUM_NODES * EMB * sizeof(float);   // 38.4 MB
  const size_t embBytesB = (size_t)NUM_BQ    * EMB * sizeof(float);   //  5.1 MB
  const size_t embBytesU = (size_t)NUM_USERS * EMB * sizeof(float);   // 25.6 MB
  char* ws = (char*)d_ws;
  float* e0     = (float*)(ws);
  float* e1     = (float*)(ws + embBytesN);
  float* bfeat  = (float*)(ws + 2 * embBytesN);
  float* deg_hv = (float*)(ws + 2 * embBytesN + embBytesB);
  float* deg_hu = deg_hv + NUM_BQ;
  float* ulocal = e0;   // e buffers are dead by the time u_local is built

  const int totalVec = NUM_NODES * (EMB / 4);   // 2.4M float4
  const int userVec  = NUM_USERS * (EMB / 4);
  const int bqVec    = NUM_BQ    * (EMB / 4);

  // 1) acc (=out) and e0 start as concat(user_emb, item_emb)
  copy_init_kernel<<<cdiv(totalVec, 256), 256, 0, stream>>>(
      user_emb, item_emb, e0, out, totalVec, userVec);

  // 2) three propagation layers: e <- A e ; acc += e (WMMA identity-accumulate)
  float* eprev = e0;
  float* enext = e1;
  for (int layer = 0; layer < 3; ++layer) {
    hipMemsetAsync(enext, 0, embBytesN, stream);
    spmm16_kernel<<<cdiv((long long)E_adj * 16, 256), 256, 0, stream>>>(
        adj_row, adj_col, adj_val, eprev, enext, nullptr, E_adj);
    acc_wmma_kernel<<<NUM_NODES / 16, 128, 0, stream>>>(enext, out);
    float* t = eprev; eprev = enext; enext = t;
  }

  // 3) biclique path (degree histograms fused into the SpMM passes)
  hipMemsetAsync(bfeat,  0, embBytesB, stream);
  hipMemsetAsync(deg_hv, 0, (size_t)NUM_BQ * sizeof(float), stream);
  hipMemsetAsync(deg_hu, 0, (size_t)NUM_USERS * sizeof(float), stream);

  spmm16_kernel<<<cdiv((long long)E_hv * 16, 256), 256, 0, stream>>>(
      hv_row, hv_col, hv_val, item_emb, bfeat, deg_hv, E_hv);
  rownorm_kernel<<<cdiv(bqVec, 256), 256, 0, stream>>>(bfeat, deg_hv, bqVec);

  hipMemsetAsync(ulocal, 0, embBytesU, stream);
  spmm16_kernel<<<cdiv((long long)E_hu * 16, 256), 256, 0, stream>>>(
      hu_row, hu_col, hu_val, bfeat, ulocal, deg_hu, E_hu);

  // 4) out = acc/4 (+ u_local / deg_hu on user rows)
  finalize_kernel<<<cdiv(totalVec, 256), 256, 0, stream>>>(
      out, ulocal, deg_hu, totalVec, userVec);
}